// MetaPredicate_67001489817855
// MI455X (gfx1250) — compile-verified
//
#include <hip/hip_runtime.h>
#include <cmath>

// MI455X / gfx1250, wave32.
// Reference: ret[b] = sum_r softmax(alpha)[r] * (isnan(mat[x[b],r]) ? 0
//                       : rule_tables[r, (int)mat[x[b],r]]);   slacks = sum(slack_params)
//
// Strategy: gather-bound kernel (mat 128MB + tables 32MB fit in 192MB L2).
// K=8 reduction done on the matrix pipe with two V_WMMA_F32_16X16X4_F32.

typedef __attribute__((ext_vector_type(2))) float v2f;
typedef __attribute__((ext_vector_type(8))) float v8f;

#define RR 8

// One-wave prep: softmax(alpha_logits) -> w[0..7] in d_ws, sum(slack) -> slack_out.
__global__ void meta_prep_kernel(const float* __restrict__ alpha,
                                 const float* __restrict__ slack,
                                 float* __restrict__ w,
                                 float* __restrict__ slack_out) {
  if (threadIdx.x == 0 && blockIdx.x == 0) {
    float mx = alpha[0];
#pragma unroll
    for (int r = 1; r < RR; ++r) mx = fmaxf(mx, alpha[r]);
    float e[RR];
    float s = 0.f;
#pragma unroll
    for (int r = 0; r < RR; ++r) { e[r] = expf(alpha[r] - mx); s += e[r]; }
    float inv = 1.f / s;
#pragma unroll
    for (int r = 0; r < RR; ++r) w[r] = e[r] * inv;
    float ss = 0.f;
#pragma unroll
    for (int r = 0; r < RR; ++r) ss += slack[r];
    *slack_out = ss;
  }
}

// Each wave handles 16 output rows.
// Lane layout for V_WMMA_F32_16X16X4_F32 (ISA 7.12.2):
//   A (16x4, MxK): lane m=lane&15 is row M=m; lane-half h=lane>>4 supplies
//                  K = {2h, 2h+1} in A-VGPRs {0,1}  (K = vgpr + 2h).
//   B (4x16, KxN): same K pattern per lane-half; we replicate w[k] across all N,
//                  so the value only depends on K.
//   D (16x16):     lane 0 holds D[0..7][0] in c[0..7]; lane 16 holds D[8..15][0].
// Two chained WMMAs (K=0..3 then K=4..7) accumulate the K=8 dot product.
__global__ __launch_bounds__(256) void meta_gather_wmma_kernel(
    const int*   __restrict__ x,     // [B]
    const float* __restrict__ mat,   // [N_DF][8]
    const float* __restrict__ tab,   // [8][M]
    const float* __restrict__ w,     // [8] softmax weights (in d_ws)
    float*       __restrict__ out,   // [B]
    int Btot, int M) {
  const int lane = threadIdx.x & 31;
  const int wave = threadIdx.x >> 5;
  const int m    = lane & 15;   // output row within the wave's 16-row tile
  const int h    = lane >> 4;   // lane-half -> supplies K = {2h,2h+1} (+4)
  const long long rowBase = ((long long)blockIdx.x * 8 + wave) * 16;
  if (rowBase >= Btot) return;

  if (rowBase + 16 <= (long long)Btot) {
    // -------- full-wave WMMA path (wave-uniform branch; EXEC all 1s) --------
    const long long b  = rowBase + m;
    const long long xr = (long long)x[b];
    const float* mrow  = mat + xr * 8;

    // Two f32 pairs of this row's mat values: k={2h,2h+1} and k={4+2h,5+2h}.
    v2f t0 = *(const v2f*)(mrow + 2 * h);       // 8B-aligned (row is 32B-aligned)
    v2f t1 = *(const v2f*)(mrow + 4 + 2 * h);
    // Matching softmax weights for this lane-half's K values.
    v2f wlo = *(const v2f*)(w + 2 * h);
    v2f whi = *(const v2f*)(w + 4 + 2 * h);

    const int k0 = 2 * h;
    v2f A0, A1;
    {
      bool v0 = (t0.x == t0.x);
      bool v1 = (t0.y == t0.y);
      int  i0 = v0 ? (int)t0.x : 0;
      int  i1 = v1 ? (int)t0.y : 0;
      float g0 = tab[(long long)(k0 + 0) * M + i0];
      float g1 = tab[(long long)(k0 + 1) * M + i1];
      A0.x = v0 ? g0 : 0.f;
      A0.y = v1 ? g1 : 0.f;
    }
    {
      bool v0 = (t1.x == t1.x);
      bool v1 = (t1.y == t1.y);
      int  i0 = v0 ? (int)t1.x : 0;
      int  i1 = v1 ? (int)t1.y : 0;
      float g0 = tab[(long long)(k0 + 4) * M + i0];
      float g1 = tab[(long long)(k0 + 5) * M + i1];
      A1.x = v0 ? g0 : 0.f;
      A1.y = v1 ? g1 : 0.f;
    }

    v8f c = {};
    // 8 args: (neg_a, A, neg_b, B, c_mod, C, reuse_a, reuse_b)
    c = __builtin_amdgcn_wmma_f32_16x16x4_f32(false, A0, false, wlo, (short)0, c,
                                              false, false);
    c = __builtin_amdgcn_wmma_f32_16x16x4_f32(false, A1, false, whi, (short)0, c,
                                              false, false);

    // Column 0 of D is the result vector: lane 0 -> rows 0..7, lane 16 -> rows 8..15.
    if (m == 0) {
      float* o = out + rowBase + (long long)h * 8;
#pragma unroll
      for (int i = 0; i < 8; ++i) o[i] = c[i];
    }
  } else {
    // -------- scalar tail (never taken for B % 128 == 0, kept for safety) ----
    const long long b = rowBase + lane;
    if (lane < 16 && b < (long long)Btot) {
      const float* mrow = mat + (long long)x[b] * 8;
      float acc = 0.f;
#pragma unroll
      for (int r = 0; r < RR; ++r) {
        float t = mrow[r];
        bool  v = (t == t);
        int   i = v ? (int)t : 0;
        float g = tab[(long long)r * M + i];
        acc += (v ? g : 0.f) * w[r];
      }
      out[b] = acc;
    }
  }
}

extern "C" void kernel_launch(void* const* d_in, const int* in_sizes, int n_in,
                              void* d_out, int out_size, void* d_ws, size_t ws_size,
                              hipStream_t stream) {
  // setup_inputs() order: x (int32), mat (f32), rule_tables (f32),
  //                       alpha_logits (f32), slack_params (f32)
  const int*   x     = (const int*)  d_in[0];
  const float* mat   = (const float*)d_in[1];
  const float* tab   = (const float*)d_in[2];
  const float* alpha = (const float*)d_in[3];
  const float* slack = (const float*)d_in[4];

  float* out = (float*)d_out;      // out_size = B + 1 (ret flat + slacks scalar)
  float* w   = (float*)d_ws;       // 8 floats of scratch for softmax weights

  const int Btot = in_sizes[0];
  const int M    = in_sizes[2] / RR;   // rule_tables flat = R*M

  meta_prep_kernel<<<1, 32, 0, stream>>>(alpha, slack, w, out + Btot);

  const int rowsPerBlock = 128;        // 8 waves * 16 rows
  const int grid = (Btot + rowsPerBlock - 1) / rowsPerBlock;
  meta_gather_wmma_kernel<<<grid, 256, 0, stream>>>(x, mat, tab, w, out, Btot, M);
}